// GriddingDistance_59407987638986
// MI455X (gfx1250) — compile-verified
//
#include <hip/hip_runtime.h>
#include <stdint.h>

// ---- problem constants (match reference) ----
#define BATCH   16
#define NPTS    32768
#define LSIDE   131
#define VVOX    (131*131*131)            // 2,248,091
#define BN      (BATCH*NPTS)             // 524,288
#define BV      (BATCH*VVOX)             // 35,969,456
// total out = 2*BV = 71,938,912 floats (< 2^31, int indexing OK)

// ---- monotone float <-> uint encoding for atomic min on floats ----
__device__ __forceinline__ unsigned fenc(float f) {
    unsigned u = __float_as_uint(f);
    return (u & 0x80000000u) ? ~u : (u | 0x80000000u);
}
__device__ __forceinline__ float fdec(unsigned e) {
    unsigned u = (e & 0x80000000u) ? (e ^ 0x80000000u) : ~e;
    return __uint_as_float(u);
}

// ---- hardware f32 atomic add (agent scope, relaxed -> global_atomic_add_f32) ----
__device__ __forceinline__ void atomicAddF(float* p, float v) {
    __hip_atomic_fetch_add(p, v, __ATOMIC_RELAXED, __HIP_MEMORY_SCOPE_AGENT);
}

// ---- CDNA5 async global->LDS path (probe-confirmed signatures:
//      b32:  (AS1 int*,  AS3 int*,  imm offset, imm cpol)
//      b128: (AS1 v4i*,  AS3 v4i*,  imm offset, imm cpol)) ----
#if defined(__has_builtin)
#if __has_builtin(__builtin_amdgcn_global_load_async_to_lds_b32) && \
    __has_builtin(__builtin_amdgcn_s_wait_asynccnt)
#define USE_ASYNC_LDS 1
#if __has_builtin(__builtin_amdgcn_global_load_async_to_lds_b128)
#define USE_ASYNC_B128 1
#endif
#endif
#endif
#ifndef USE_ASYNC_LDS
#define USE_ASYNC_LDS 0
#endif
#ifndef USE_ASYNC_B128
#define USE_ASYNC_B128 0
#endif

#if USE_ASYNC_LDS
typedef __attribute__((address_space(1))) int GASI;   // global int
typedef __attribute__((address_space(3))) int LASI;   // LDS int
#endif
#if USE_ASYNC_B128
typedef int v4i_t __attribute__((__vector_size__(4 * sizeof(int))));
typedef __attribute__((address_space(1))) v4i_t GASV4;   // global int4
typedef __attribute__((address_space(3))) v4i_t LASV4;   // LDS int4
#endif

// ============================ kernels ============================
// (scatter first so its assembly shows in the disasm snippet)

// One point per thread. 4096 blocks x 256 threads == 2*BN exactly; BN = 2048*256,
// so every block lives entirely inside one cloud.
__global__ void __launch_bounds__(256)
Gridding_scatter(const float* __restrict__ pred,
                 const float* __restrict__ gt,
                 const unsigned* __restrict__ gmins,
                 float* __restrict__ out) {
    __shared__ __align__(16) float spts[256 * 3];

    const int t = threadIdx.x;
    const int g0 = blockIdx.x * 256;           // first point of block
    const int cloud = (g0 >= BN) ? 1 : 0;      // uniform per block
    const int q0 = g0 - cloud * BN;
    const int q = q0 + t;
    const float* src = cloud ? gt : pred;

#if USE_ASYNC_B128
    {
        // Cooperative async DMA: this block's 256 points = 3072 contiguous,
        // 16B-aligned bytes. Lanes 0..191 each move 16B via async B128.
        if (t < 192) {
            GASV4* ga = (GASV4*)((uintptr_t)(src + (size_t)q0 * 3) + (uintptr_t)(t * 16));
            LASV4* la = (LASV4*)((uintptr_t)spts + (uintptr_t)(t * 16));
            __builtin_amdgcn_global_load_async_to_lds_b128(ga, la, 0, 0);
        }
        __builtin_amdgcn_s_wait_asynccnt(0);   // each wave drains its own asyncs
        __syncthreads();                        // then all LDS data is visible
    }
#elif USE_ASYNC_LDS
    {
        GASI* ga = (GASI*)(uintptr_t)(src + (size_t)q * 3);
        LASI* la = (LASI*)(uintptr_t)(&spts[t * 3]);
        __builtin_amdgcn_global_load_async_to_lds_b32(ga, la, 0, 0);
        __builtin_amdgcn_global_load_async_to_lds_b32(ga, la, 4, 0);
        __builtin_amdgcn_global_load_async_to_lds_b32(ga, la, 8, 0);
        __builtin_amdgcn_s_wait_asynccnt(0);   // lanes read only their own slots
    }
#else
    {
        size_t o = (size_t)q * 3;
        spts[t * 3 + 0] = src[o + 0];
        spts[t * 3 + 1] = src[o + 1];
        spts[t * 3 + 2] = src[o + 2];
        __syncthreads();
    }
#endif

    // min corner: floor(min) - 1, per axis (integer-valued)
    const int mx = (int)(floorf(fdec(gmins[0])) - 1.0f);
    const int my = (int)(floorf(fdec(gmins[1])) - 1.0f);
    const int mz = (int)(floorf(fdec(gmins[2])) - 1.0f);

    const float sx = spts[t * 3 + 0] * 64.0f;
    const float sy = spts[t * 3 + 1] * 64.0f;
    const float sz = spts[t * 3 + 2] * 64.0f;

    const float mask = (((sx + sy) + sz) != 0.0f) ? 1.0f : 0.0f;

    const float fx = floorf(sx), fy = floorf(sy), fz = floorf(sz);
    const float rx = sx - fx, ry = sy - fy, rz = sz - fz;

    const int ix = (int)fx - mx;
    const int iy = (int)fy - my;
    const int iz = (int)fz - mz;

    const int b = q >> 15;   // q / NPTS
    const int base = cloud * BV + b * VVOX + (ix * LSIDE + iy) * LSIDE + iz;

    const float wx0 = 1.0f - rx, wx1 = rx;
    const float wy0 = 1.0f - ry, wy1 = ry;
    const float wz0 = 1.0f - rz, wz1 = rz;

    // OFFSETS order: (0,0,0)(0,0,1)(0,1,0)(0,1,1)(1,0,0)(1,0,1)(1,1,0)(1,1,1)
    const float w000 = ((wx0 * wy0) * wz0) * mask;
    const float w001 = ((wx0 * wy0) * wz1) * mask;
    const float w010 = ((wx0 * wy1) * wz0) * mask;
    const float w011 = ((wx0 * wy1) * wz1) * mask;
    const float w100 = ((wx1 * wy0) * wz0) * mask;
    const float w101 = ((wx1 * wy0) * wz1) * mask;
    const float w110 = ((wx1 * wy1) * wz0) * mask;
    const float w111 = ((wx1 * wy1) * wz1) * mask;

    float* o = out + base;
    atomicAddF(o + 0,                 w000);
    atomicAddF(o + 1,                 w001);
    atomicAddF(o + LSIDE,             w010);
    atomicAddF(o + LSIDE + 1,         w011);
    atomicAddF(o + LSIDE * LSIDE,     w100);
    atomicAddF(o + LSIDE * LSIDE + 1, w101);
    atomicAddF(o + LSIDE * LSIDE + LSIDE,     w110);
    atomicAddF(o + LSIDE * LSIDE + LSIDE + 1, w111);
}

__global__ void Gridding_init(unsigned* __restrict__ gmins) {
    if (threadIdx.x < 3) gmins[threadIdx.x] = 0xFFFFFFFFu;   // encoded +inf
}

// Joint per-axis min over BOTH scaled clouds (reference takes min before masking).
__global__ void Gridding_minreduce(const float* __restrict__ pred,
                                   const float* __restrict__ gt,
                                   unsigned* __restrict__ gmins) {
    __shared__ unsigned smin[3];
    if (threadIdx.x < 3) smin[threadIdx.x] = 0xFFFFFFFFu;
    __syncthreads();

    unsigned m0 = 0xFFFFFFFFu, m1 = 0xFFFFFFFFu, m2 = 0xFFFFFFFFu;
    const int total = 2 * BN;
    for (int i = blockIdx.x * blockDim.x + threadIdx.x; i < total;
         i += gridDim.x * blockDim.x) {
        const float* src = (i < BN) ? pred : gt;
        int q = (i < BN) ? i : (i - BN);
        size_t o = (size_t)q * 3;
        float x = src[o + 0] * 64.0f;
        float y = src[o + 1] * 64.0f;
        float z = src[o + 2] * 64.0f;
        m0 = min(m0, fenc(x));
        m1 = min(m1, fenc(y));
        m2 = min(m2, fenc(z));
    }
    atomicMin(&smin[0], m0);   // ds_min_u32
    atomicMin(&smin[1], m1);
    atomicMin(&smin[2], m2);
    __syncthreads();
    if (threadIdx.x < 3) atomicMin(&gmins[threadIdx.x], smin[threadIdx.x]);
}

// Output must be re-zeroed on every call (graph replays would accumulate).
__global__ void Gridding_zero(float4* __restrict__ out4, int n4,
                              float* __restrict__ out, int ntail) {
    int stride = gridDim.x * blockDim.x;
    for (int i = blockIdx.x * blockDim.x + threadIdx.x; i < n4; i += stride)
        out4[i] = make_float4(0.f, 0.f, 0.f, 0.f);
    if (blockIdx.x == 0 && threadIdx.x < (unsigned)ntail)
        out[4 * n4 + threadIdx.x] = 0.f;
}

// ============================ launcher ============================

extern "C" void kernel_launch(void* const* d_in, const int* in_sizes, int n_in,
                              void* d_out, int out_size, void* d_ws, size_t ws_size,
                              hipStream_t stream) {
    const float* pred = (const float*)d_in[0];
    const float* gt   = (const float*)d_in[1];
    float* out        = (float*)d_out;
    unsigned* gmins   = (unsigned*)d_ws;

    Gridding_init<<<1, 64, 0, stream>>>(gmins);
    Gridding_minreduce<<<1024, 256, 0, stream>>>(pred, gt, gmins);

    int n4 = out_size / 4;
    int ntail = out_size - 4 * n4;   // 0 for this problem, handled anyway
    Gridding_zero<<<2048, 256, 0, stream>>>((float4*)out, n4, out, ntail);

    Gridding_scatter<<<(2 * BN) / 256, 256, 0, stream>>>(pred, gt, gmins, out);
}